// TransformerSubsetAggregator_6167573037355
// MI455X (gfx1250) — compile-verified
//
#include <hip/hip_runtime.h>
#include <hip/hip_bf16.h>
#include <math.h>

// Problem constants (match reference)
#define NCHAINS 8192
#define LSEQ    128
#define DIM     256
#define KSEL    8
#define NHEAD   4
#define DHEAD   64
#define FFD     1024

// LDS strides padded so 16-lane WMMA operand fetches are bank-conflict free
// (stride % 64 == 4 -> lane ln hits banks {4ln, 4ln+1}; upper half {4ln+2, 4ln+3})
#define XS 260    // padded stride for 256-wide slabs
#define QS 772    // padded stride for 768-wide qkv slab
#define GS 1028   // padded stride for 1024-wide ff/gelu slab

typedef float v2f __attribute__((ext_vector_type(2)));
typedef float v8f __attribute__((ext_vector_type(8)));

__device__ __forceinline__ v8f wmma4(v2f a, v2f b, v8f c) {
  // D = A(16x4 f32) * B(4x16 f32) + C(16x16 f32)
  return __builtin_amdgcn_wmma_f32_16x16x4_f32(false, a, false, b, (short)0, c,
                                               false, false);
}

__device__ __forceinline__ float gelu_exact(float x) {
  return 0.5f * x * (1.0f + erff(x * 0.70710678118654752440f));
}

// Two stacked 16-row M tiles (rows 0..15 and 16..31 of A) x one 16-col N tile.
// A: LDS f32 row-major stride sa.  B: global, B[k][n] = W[n][k], W row-major
// stride sb (so the per-lane fetch is two consecutive k from row n0+ln).
__device__ __forceinline__ void gemm_tile2(const float* __restrict__ A, int sa,
                                           const float* __restrict__ W, int sb,
                                           int n0, int K, int lane, bool alignedB,
                                           v8f& acc0, v8f& acc1) {
  const int ln = lane & 15;
  const int kh = (lane >> 4) << 1;   // 0 or 2
  const float* a0 = A + ln * sa + kh;
  const float* a1 = A + (16 + ln) * sa + kh;
  const float* bp = W + (size_t)(n0 + ln) * sb + kh;
  for (int k = 0; k < K; k += 4) {
    v2f b;
    if (alignedB) {
      b = *(const v2f*)(bp + k);
    } else {
      b.x = bp[k]; b.y = bp[k + 1];
    }
    v2f a = *(const v2f*)(a0 + k);
    acc0 = wmma4(a, b, acc0);
    a = *(const v2f*)(a1 + k);
    acc1 = wmma4(a, b, acc1);
  }
}

// LayerNorm over 32 rows x 256 cols held in LDS (stride XS). 8 threads/row.
__device__ __forceinline__ void lnorm32(const float* __restrict__ src,
                                        float* __restrict__ dst,
                                        const float* __restrict__ g,
                                        const float* __restrict__ b,
                                        float* red, int tid) {
  const int r = tid >> 3, s = tid & 7;
  const float* x = src + r * XS + s * 32;
  float sum = 0.f;
#pragma unroll
  for (int i = 0; i < 32; ++i) sum += x[i];
  red[tid] = sum;
  __syncthreads();
  float mu = 0.f;
#pragma unroll
  for (int i = 0; i < 8; ++i) mu += red[(r << 3) + i];
  mu *= (1.0f / 256.0f);
  __syncthreads();
  float vs = 0.f;
#pragma unroll
  for (int i = 0; i < 32; ++i) { float d = x[i] - mu; vs += d * d; }
  red[tid] = vs;
  __syncthreads();
  float var = 0.f;
#pragma unroll
  for (int i = 0; i < 8; ++i) var += red[(r << 3) + i];
  var *= (1.0f / 256.0f);
  const float rs = rsqrtf(var + 1e-5f);
  float* y = dst + r * XS + s * 32;
  const float* gg = g + s * 32;
  const float* bb = b + s * 32;
#pragma unroll
  for (int i = 0; i < 32; ++i) y[i] = (x[i] - mu) * rs * gg[i] + bb[i];
  __syncthreads();
}

// ---------------------------------------------------------------------------
// Kernel 1: stable top-8 masked selection per chain; indices sorted ascending;
// sel[c][j] = v-row index (batch_idx gathered) or -1 if slot not picked.
// ---------------------------------------------------------------------------
__global__ __launch_bounds__(256)
void tsa_topk_sel(const float* __restrict__ rank, const unsigned char* __restrict__ mask,
                  const int* __restrict__ bidx, int* __restrict__ sel) {
  const int c = blockIdx.x * blockDim.x + threadIdx.x;
  if (c >= NCHAINS) return;
  const float* sc = rank + (size_t)c * LSEQ;
  const unsigned char* mk = mask + (size_t)c * LSEQ;
  float bv[KSEL];
  int   bi[KSEL];
  int cnt = 0;
  for (int i = 0; i < LSEQ; ++i) {
    if (!mk[i]) continue;
    const float s = sc[i];
    if (cnt == KSEL && !(s > bv[KSEL - 1])) continue;
    const int n = (cnt < KSEL) ? cnt : KSEL;
    int p = (cnt < KSEL) ? cnt : (KSEL - 1);
    for (int j = 0; j < n; ++j) {
      if (s > bv[j]) { p = j; break; }
    }
    const int last = (cnt < KSEL) ? cnt : (KSEL - 1);
    for (int j = last; j > p; --j) { bv[j] = bv[j - 1]; bi[j] = bi[j - 1]; }
    bv[p] = s; bi[p] = i;
    if (cnt < KSEL) ++cnt;
  }
  // sort the picked indices ascending (matches sort-with-sentinel in reference)
  for (int a = 1; a < cnt; ++a) {
    const int key = bi[a];
    int b = a - 1;
    while (b >= 0 && bi[b] > key) { bi[b + 1] = bi[b]; --b; }
    bi[b + 1] = key;
  }
  const int* br = bidx + (size_t)c * LSEQ;
  for (int j = 0; j < KSEL; ++j)
    sel[(size_t)c * KSEL + j] = (j < cnt) ? br[bi[j]] : -1;
}

// ---------------------------------------------------------------------------
// Kernel 2: x[c,k,:] = (picked ? v[member] : 0) + pos_embed[k,:]
// 4 rows / block of 256 threads; float4 vectorized.
// ---------------------------------------------------------------------------
__global__ __launch_bounds__(256)
void tsa_build_x(const int* __restrict__ sel, const float* __restrict__ v,
                 const float* __restrict__ pe, float* __restrict__ x) {
  const int tid = threadIdx.x;
  const int r = blockIdx.x * 4 + (tid >> 6);    // global row in [0, 65536)
  const int lc = tid & 63;                      // float4 index within row
  const int kk = r & (KSEL - 1);
  const int m = sel[r];
  float4 p = ((const float4*)(pe + kk * DIM))[lc];
  float4 val = make_float4(0.f, 0.f, 0.f, 0.f);
  if (m >= 0) val = ((const float4*)(v + (size_t)m * DIM))[lc];
  float4 o;
  o.x = val.x + p.x; o.y = val.y + p.y; o.z = val.z + p.z; o.w = val.w + p.w;
  ((float4*)(x + (size_t)r * DIM))[lc] = o;
}

// ---------------------------------------------------------------------------
// Kernel 3: encoder layer + masked mean pool.  4 chains (32 rows) / workgroup,
// 256 threads = 8 wave32.  All GEMMs via v_wmma_f32_16x16x4_f32.
// ---------------------------------------------------------------------------
__global__ __launch_bounds__(256)
void tsa_encoder(const int* __restrict__ sel, const float* __restrict__ xg,
                 float* __restrict__ pooled,
                 const float* __restrict__ w_qkv, const float* __restrict__ b_qkv,
                 const float* __restrict__ w_o,   const float* __restrict__ b_o,
                 const float* __restrict__ ln1g,  const float* __restrict__ ln1b,
                 const float* __restrict__ ln2g,  const float* __restrict__ ln2b,
                 const float* __restrict__ w_ff1, const float* __restrict__ b_ff1,
                 const float* __restrict__ w_ff2, const float* __restrict__ b_ff2) {
  extern __shared__ float sm[];
  float* X   = sm;                 // 32 x XS
  float* H   = X + 32 * XS;        // 32 x XS
  float* BIG = H + 32 * XS;        // 32 x GS  (qkv phase uses stride QS)
  float* RED = BIG + 32 * GS;      // 256
  float* PK  = RED + 256;          // 32
  float* DEN = PK + 32;            // 4
  float* ANY = DEN + 4;            // 4

  const int tid = threadIdx.x;
  const int lane = tid & 31;
  const int wv = tid >> 5;
  const int c0 = blockIdx.x * 4;
  const int row0 = c0 * KSEL;      // first global row of this block

  // Load x tile (rows are 256 floats; padded LDS rows are 65 float4)
  const float4* xg4 = (const float4*)(xg + (size_t)row0 * DIM);
  float4* X4 = (float4*)X;
  for (int i = tid; i < 32 * 64; i += 256) {
    const int r = i >> 6, cc = i & 63;
    X4[r * (XS / 4) + cc] = xg4[i];
  }
  if (tid < 32) PK[tid] = (sel[row0 + tid] >= 0) ? 1.f : 0.f;
  __syncthreads();
  if (tid < 4) {
    float cnt = 0.f;
#pragma unroll
    for (int k = 0; k < KSEL; ++k) cnt += PK[tid * KSEL + k];
    ANY[tid] = (cnt > 0.f) ? 1.f : 0.f;
    DEN[tid] = fmaxf(cnt, 1e-8f);
  }
  __syncthreads();

  // --- LN1: X -> H ---
  lnorm32(X, H, ln1g, ln1b, RED, tid);

  // --- qkv = H @ w_qkv^T + b_qkv  -> BIG (stride QS, cols 0..767) ---
  for (int nt = wv; nt < 48; nt += 8) {
    const int n0 = nt * 16;
    v8f a0 = {}, a1 = {};
    gemm_tile2(H, XS, w_qkv, DIM, n0, DIM, lane, true, a0, a1);
    const int ln = lane & 15;
    const int mh = (lane >> 4) << 3;
    const float bv = b_qkv[n0 + ln];
#pragma unroll
    for (int r = 0; r < 8; ++r) {
      BIG[(r + mh) * QS + n0 + ln]      = a0[r] + bv;
      BIG[(16 + r + mh) * QS + n0 + ln] = a1[r] + bv;
    }
  }
  __syncthreads();

  // --- attention (K=8, tiny): VALU, exact softmax, kpm + empty-chain fix ---
  if (tid < 128) {
    const int cid = tid >> 5, rem = tid & 31;
    const int hh = rem >> 3, qi = rem & 7;
    const int rb = cid * KSEL;
    const bool anyp = ANY[cid] > 0.f;
    const float* q = BIG + (rb + qi) * QS + hh * DHEAD;
    float scv[KSEL];
    float mx = -3.4e38f;
#pragma unroll
    for (int k = 0; k < KSEL; ++k) {
      const bool allow = (PK[rb + k] > 0.f) || (!anyp && k == 0);
      float s = -3.4e38f;
      if (allow) {
        const float* kp = BIG + (rb + k) * QS + DIM + hh * DHEAD;
        float d = 0.f;
        for (int t = 0; t < DHEAD; ++t) d += q[t] * kp[t];
        s = d * 0.125f;   // 1/sqrt(64)
      }
      scv[k] = s;
      mx = fmaxf(mx, s);
    }
    float ssum = 0.f;
#pragma unroll
    for (int k = 0; k < KSEL; ++k) { const float e = expf(scv[k] - mx); scv[k] = e; ssum += e; }
    const float inv = 1.f / ssum;
    float o[DHEAD];
    for (int t = 0; t < DHEAD; ++t) o[t] = 0.f;
#pragma unroll
    for (int k = 0; k < KSEL; ++k) {
      const float w = scv[k] * inv;
      const float* vp = BIG + (rb + k) * QS + 2 * DIM + hh * DHEAD;
      for (int t = 0; t < DHEAD; ++t) o[t] += w * vp[t];
    }
    float* op = H + (rb + qi) * XS + hh * DHEAD;   // H now holds attn output
    for (int t = 0; t < DHEAD; ++t) op[t] = o[t];
  }
  __syncthreads();

  // --- x += H @ w_o^T + b_o ---
  for (int nt = wv; nt < 16; nt += 8) {
    const int n0 = nt * 16;
    v8f a0 = {}, a1 = {};
    gemm_tile2(H, XS, w_o, DIM, n0, DIM, lane, true, a0, a1);
    const int ln = lane & 15;
    const int mh = (lane >> 4) << 3;
    const float bv = b_o[n0 + ln];
#pragma unroll
    for (int r = 0; r < 8; ++r) {
      X[(r + mh) * XS + n0 + ln]      += a0[r] + bv;
      X[(16 + r + mh) * XS + n0 + ln] += a1[r] + bv;
    }
  }
  __syncthreads();

  // --- LN2: X -> H ---
  lnorm32(X, H, ln2g, ln2b, RED, tid);

  // --- ff1: BIG(stride GS) = gelu(H @ w_ff1^T + b_ff1) ---
  for (int nt = wv; nt < 64; nt += 8) {
    const int n0 = nt * 16;
    v8f a0 = {}, a1 = {};
    gemm_tile2(H, XS, w_ff1, DIM, n0, DIM, lane, true, a0, a1);
    const int ln = lane & 15;
    const int mh = (lane >> 4) << 3;
    const float bv = b_ff1[n0 + ln];
#pragma unroll
    for (int r = 0; r < 8; ++r) {
      BIG[(r + mh) * GS + n0 + ln]      = gelu_exact(a0[r] + bv);
      BIG[(16 + r + mh) * GS + n0 + ln] = gelu_exact(a1[r] + bv);
    }
  }
  __syncthreads();

  // --- x += BIG @ w_ff2^T + b_ff2 ---
  for (int nt = wv; nt < 16; nt += 8) {
    const int n0 = nt * 16;
    v8f a0 = {}, a1 = {};
    gemm_tile2(BIG, GS, w_ff2, FFD, n0, FFD, lane, true, a0, a1);
    const int ln = lane & 15;
    const int mh = (lane >> 4) << 3;
    const float bv = b_ff2[n0 + ln];
#pragma unroll
    for (int r = 0; r < 8; ++r) {
      X[(r + mh) * XS + n0 + ln]      += a0[r] + bv;
      X[(16 + r + mh) * XS + n0 + ln] += a1[r] + bv;
    }
  }
  __syncthreads();

  // --- masked mean pool ---
  for (int i = tid; i < 4 * DIM; i += 256) {
    const int cid = i >> 8;
    const int d = i & 255;
    float s = 0.f;
#pragma unroll
    for (int k = 0; k < KSEL; ++k) s += PK[cid * KSEL + k] * X[(cid * KSEL + k) * XS + d];
    const float res = (ANY[cid] > 0.f) ? (s / DEN[cid]) : 0.f;
    pooled[(size_t)(c0 + cid) * DIM + d] = res;
  }
}

// ---------------------------------------------------------------------------
// Kernel 4: MLP head over [pooled | log1p(count)].  32 chains / workgroup.
// ---------------------------------------------------------------------------
__global__ __launch_bounds__(256)
void tsa_head(const float* __restrict__ pooled, const float* __restrict__ count,
              const float* __restrict__ w1, const float* __restrict__ b1,
              const float* __restrict__ w2, const float* __restrict__ b2,
              float* __restrict__ out) {
  extern __shared__ float sm[];
  float* P  = sm;                 // 32 x XS
  float* G  = P + 32 * XS;        // 32 x GS
  float* LC = G + 32 * GS;        // 32

  const int tid = threadIdx.x;
  const int lane = tid & 31;
  const int wv = tid >> 5;
  const int r0 = blockIdx.x * 32;

  const float4* pg = (const float4*)(pooled + (size_t)r0 * DIM);
  float4* P4 = (float4*)P;
  for (int i = tid; i < 32 * 64; i += 256) {
    const int r = i >> 6, cc = i & 63;
    P4[r * (XS / 4) + cc] = pg[i];
  }
  if (tid < 32) LC[tid] = log1pf(count[r0 + tid]);
  __syncthreads();

  // G = gelu(feat @ w1^T + b1)  with feat = [pooled | lc]; w1 is [1024][257]
  for (int nt = wv; nt < 64; nt += 8) {
    const int n0 = nt * 16;
    v8f a0 = {}, a1 = {};
    gemm_tile2(P, XS, w1, DIM + 1, n0, DIM, lane, false, a0, a1);
    const int ln = lane & 15;
    const int mh = (lane >> 4) << 3;
    const float bv = b1[n0 + ln];
    const float wl = w1[(size_t)(n0 + ln) * (DIM + 1) + DIM];  // 257th column
#pragma unroll
    for (int r = 0; r < 8; ++r) {
      const int m0 = r + mh;
      const int m1 = 16 + r + mh;
      G[m0 * GS + n0 + ln] = gelu_exact(a0[r] + bv + LC[m0] * wl);
      G[m1 * GS + n0 + ln] = gelu_exact(a1[r] + bv + LC[m1] * wl);
    }
  }
  __syncthreads();

  // out = G @ w2^T + b2
  for (int nt = wv; nt < 16; nt += 8) {
    const int n0 = nt * 16;
    v8f a0 = {}, a1 = {};
    gemm_tile2(G, GS, w2, FFD, n0, FFD, lane, true, a0, a1);
    const int ln = lane & 15;
    const int mh = (lane >> 4) << 3;
    const float bv = b2[n0 + ln];
#pragma unroll
    for (int r = 0; r < 8; ++r) {
      out[(size_t)(r0 + r + mh) * DIM + n0 + ln]      = a0[r] + bv;
      out[(size_t)(r0 + 16 + r + mh) * DIM + n0 + ln] = a1[r] + bv;
    }
  }
}

// ---------------------------------------------------------------------------
extern "C" void kernel_launch(void* const* d_in, const int* in_sizes, int n_in,
                              void* d_out, int out_size, void* d_ws, size_t ws_size,
                              hipStream_t stream) {
  const float*         v        = (const float*)d_in[0];
  const int*           batchidx = (const int*)d_in[1];
  const unsigned char* mask     = (const unsigned char*)d_in[2];
  const float*         count    = (const float*)d_in[3];
  const float*         rank     = (const float*)d_in[4];
  const float*         pe       = (const float*)d_in[5];
  const float*         w_qkv    = (const float*)d_in[6];
  const float*         b_qkv    = (const float*)d_in[7];
  const float*         w_o      = (const float*)d_in[8];
  const float*         b_o      = (const float*)d_in[9];
  const float*         ln1g     = (const float*)d_in[10];
  const float*         ln1b     = (const float*)d_in[11];
  const float*         ln2g     = (const float*)d_in[12];
  const float*         ln2b     = (const float*)d_in[13];
  const float*         w_ff1    = (const float*)d_in[14];
  const float*         b_ff1    = (const float*)d_in[15];
  const float*         w_ff2    = (const float*)d_in[16];
  const float*         b_ff2    = (const float*)d_in[17];
  const float*         w_out1   = (const float*)d_in[18];
  const float*         b_out1   = (const float*)d_in[19];
  const float*         w_out2   = (const float*)d_in[20];
  const float*         b_out2   = (const float*)d_in[21];

  // Workspace carve (all offsets 256B-aligned):
  //   sel    : 8192*8 int      =   256 KB
  //   x      : 8192*8*256 f32  =  64   MB
  //   pooled : 8192*256 f32    =   8   MB
  char* ws = (char*)d_ws;
  int*   sel    = (int*)ws;
  float* x      = (float*)(ws + (size_t)NCHAINS * KSEL * sizeof(int));
  float* pooled = (float*)(ws + (size_t)NCHAINS * KSEL * sizeof(int)
                              + (size_t)NCHAINS * KSEL * DIM * sizeof(float));

  constexpr int SMB = (32 * XS * 2 + 32 * GS + 256 + 32 + 4 + 4) * (int)sizeof(float);
  constexpr int SMC = (32 * XS + 32 * GS + 32) * (int)sizeof(float);

  tsa_topk_sel<<<NCHAINS / 256, 256, 0, stream>>>(rank, mask, batchidx, sel);
  tsa_build_x<<<(NCHAINS * KSEL) / 4, 256, 0, stream>>>(sel, v, pe, x);
  tsa_encoder<<<NCHAINS / 4, 256, SMB, stream>>>(sel, x, pooled,
                                                 w_qkv, b_qkv, w_o, b_o,
                                                 ln1g, ln1b, ln2g, ln2b,
                                                 w_ff1, b_ff1, w_ff2, b_ff2);
  tsa_head<<<NCHAINS / 32, 256, SMC, stream>>>(pooled, count, w_out1, b_out1,
                                               w_out2, b_out2, (float*)d_out);
}